// CompetitiveLayer_3839700762667
// MI455X (gfx1250) — compile-verified
//
#include <hip/hip_runtime.h>

typedef __attribute__((ext_vector_type(2))) float v2f;
typedef __attribute__((ext_vector_type(4))) float v4f;
typedef __attribute__((ext_vector_type(8))) float v8f;

#define NDIM 4096

// ---------------------------------------------------------------------------
// AFout[r] = AT[r] / (1 + sum_j P[r][j]^2 * x[j])       (y = K @ x, row GEMV)
// Block b owns rows [16b, 16b+16). Wave w covers j in [512w, 512w+512).
// WMMA f32 16x16x4: A = 16x4 tile of K (squared param), B = x chunk
// replicated across all 16 columns -> every column of D holds the same 16
// row-partials; extract column 0 (lanes 0 and 16), combine 8 wave partials
// in LDS in fixed order (deterministic).
// ---------------------------------------------------------------------------
__global__ void __launch_bounds__(256) row_solve_kernel(
    const float* __restrict__ P, const float* __restrict__ AT,
    const float* __restrict__ x, float* __restrict__ AFout) {
  __shared__ float sred[128];
  const int tid  = threadIdx.x;
  const int lane = tid & 31;
  const int wave = tid >> 5;
  const int m    = lane & 15;
  const int hi   = lane >> 4;           // 0: lanes 0-15, 1: lanes 16-31
  const int r0   = blockIdx.x << 4;     // 16 rows per block
  const int jb   = wave << 9;           // 512 K-columns per wave

  // A tile: lane holds K[r0+m][j+2*hi], K[r0+m][j+2*hi+1]  (b64 load)
  const float* Arow = P + (size_t)(r0 + m) * NDIM + 2 * hi;

  v8f acc = {};
  int j = jb;
#pragma unroll 4
  for (int s = 0; s < 128; ++s, j += 4) {
    v2f p2 = *(const v2f*)(Arow + j);
    v2f a;
    a.x = p2.x * p2.x;                  // K = param^2, squared on the fly
    a.y = p2.y * p2.y;
    // B layout: VGPR0 = rows k=0/1, VGPR1 = rows k=2/3 (broadcast loads)
    v2f b;
    b.x = x[j + hi];
    b.y = x[j + hi + 2];
    acc = __builtin_amdgcn_wmma_f32_16x16x4_f32(
        false, a, false, b, (short)0, acc, false, false);
  }

  // D column 0: lane 0 holds M=0..7 in acc[0..7], lane 16 holds M=8..15
  if (m == 0) {
    const int rb = hi << 3;
#pragma unroll
    for (int v = 0; v < 8; ++v) sred[wave * 16 + rb + v] = acc[v];
  }
  __syncthreads();
  if (tid < 16) {
    float y = 0.f;
#pragma unroll
    for (int w = 0; w < 8; ++w) y += sred[w * 16 + tid];   // fixed order
    const int r = r0 + tid;
    AFout[r] = AT[r] / (1.0f + y);
  }
}

// ---------------------------------------------------------------------------
// BFout[c] = BT[c] / (1 + sum_i AF[i] * P[i][c]^2)     (z = K^T @ AF)
// Block b owns columns [16b, 16b+16). Wave w covers i in [512w, 512w+512).
// Roles swapped: B = 4x16 row-major K tile (fully coalesced 64B loads per
// half-wave), A = AF chunk replicated across all 16 rows -> every row of D
// holds the 16 column-partials; extract row 0 (VGPR0 of lanes 0-15).
// ---------------------------------------------------------------------------
__global__ void __launch_bounds__(256) col_solve_kernel(
    const float* __restrict__ P, const float* __restrict__ BT,
    const float* __restrict__ AF, float* __restrict__ BFout) {
  __shared__ float sred[128];
  const int tid  = threadIdx.x;
  const int lane = tid & 31;
  const int wave = tid >> 5;
  const int m    = lane & 15;
  const int hi   = lane >> 4;
  const int c0   = blockIdx.x << 4;     // 16 columns per block
  const int ib   = wave << 9;           // 512 K-rows per wave

  // B tile: VGPR0 <- rows i+hi, VGPR1 <- rows i+2+hi, column c0+m
  const float* Bp0 = P + (size_t)(ib + hi) * NDIM + c0 + m;
  const float* Bp1 = Bp0 + (size_t)2 * NDIM;
  const float* Ap  = AF + ib + 2 * hi;  // replicated AF chunk (b64 broadcast)

  v8f acc = {};
#pragma unroll 4
  for (int s = 0; s < 128; ++s) {
    float q0 = Bp0[0];
    float q1 = Bp1[0];
    v2f b;
    b.x = q0 * q0;
    b.y = q1 * q1;
    v2f a = *(const v2f*)Ap;
    acc = __builtin_amdgcn_wmma_f32_16x16x4_f32(
        false, a, false, b, (short)0, acc, false, false);
    Bp0 += (size_t)4 * NDIM;
    Bp1 += (size_t)4 * NDIM;
    Ap  += 4;
  }

  // D row M=0: VGPR0 (acc[0]) of lanes 0-15, N = m
  if (hi == 0) sred[wave * 16 + m] = acc[0];
  __syncthreads();
  if (tid < 16) {
    float z = 0.f;
#pragma unroll
    for (int w = 0; w < 8; ++w) z += sred[w * 16 + tid];   // fixed order
    const int c = c0 + tid;
    BFout[c] = BT[c] / (1.0f + z);
  }
}

// ---------------------------------------------------------------------------
// C[i][j] = P[i][j]^2 * AF[i] * BF[j]   — float4-vectorized streaming pass.
// ---------------------------------------------------------------------------
__global__ void __launch_bounds__(256) outer_kernel(
    const float* __restrict__ P, const float* __restrict__ AF,
    const float* __restrict__ BF, float* __restrict__ C) {
  const size_t idx4 = (((size_t)blockIdx.x * blockDim.x) + threadIdx.x) * 4;
  const int i = (int)(idx4 >> 12);       // row (NDIM = 4096)
  const int j = (int)(idx4 & (NDIM - 1));
  v4f p  = *(const v4f*)(P + idx4);
  v4f bf = *(const v4f*)(BF + j);
  float af = AF[i];
  v4f r = p * p * af * bf;
  *(v4f*)(C + idx4) = r;
}

extern "C" void kernel_launch(void* const* d_in, const int* in_sizes, int n_in,
                              void* d_out, int out_size, void* d_ws, size_t ws_size,
                              hipStream_t stream) {
  (void)in_sizes; (void)n_in; (void)out_size; (void)ws_size;
  const float* AT = (const float*)d_in[0];
  const float* BT = (const float*)d_in[1];
  const float* P  = (const float*)d_in[2];
  float* AF = (float*)d_ws;          // 4096 floats
  float* BF = AF + NDIM;             // 4096 floats  (32 KB total scratch)
  float* C  = (float*)d_out;

  // 20 scan iterations + 1 "last iterate" pass = 21 of each half-step.
  // First row-solve consumes BT directly (initial carry), so no init kernel.
  for (int it = 0; it < 21; ++it) {
    const float* x = (it == 0) ? BT : BF;
    row_solve_kernel<<<NDIM / 16, 256, 0, stream>>>(P, AT, x, AF);
    col_solve_kernel<<<NDIM / 16, 256, 0, stream>>>(P, BT, AF, BF);
  }
  outer_kernel<<<(NDIM * (NDIM / 4)) / 256, 256, 0, stream>>>(P, AF, BF, C);
}